// Detect_40870908788918
// MI455X (gfx1250) — compile-verified
//
#include <hip/hip_runtime.h>
#include <cstdint>
#include <cstddef>

#define B_     16
#define P_     32768
#define C_     41
#define TOPK_  64
#define NTH    1024
#define K_     (P_ / NTH)      // 32 scores per thread
#define CONF_T_ 0.6f
#define NMS_T_  0.5f
#define NEGF   (-__builtin_inff())
#define IMAX   0x7FFFFFFF

// ---- CDNA5 async global->LDS path (gfx1250) -------------------------------
#if defined(__gfx1250__) && __has_builtin(__builtin_amdgcn_global_load_async_to_lds_b128)
#define USE_ASYNC 1
#else
#define USE_ASYNC 0
#if defined(__gfx1250__)
#pragma message("gfx1250: global_load_async_to_lds_b128 builtin NOT available; using direct-load fallback")
#endif
#endif

#if USE_ASYNC
typedef int v4i __attribute__((ext_vector_type(4)));
typedef __attribute__((address_space(1))) v4i as1_v4i;   // global
typedef __attribute__((address_space(3))) v4i as3_v4i;   // LDS

__device__ __forceinline__ void async_b128(const float4* g, float4* l) {
  __builtin_amdgcn_global_load_async_to_lds_b128((as1_v4i*)g, (as3_v4i*)l, 0, 0);
}
#if __has_builtin(__builtin_amdgcn_s_wait_asynccnt)
#define WAIT_ASYNC_1() __builtin_amdgcn_s_wait_asynccnt(1)
#define WAIT_ASYNC_0() __builtin_amdgcn_s_wait_asynccnt(0)
#else
#define WAIT_ASYNC_1() asm volatile("s_wait_asynccnt 1" ::: "memory")
#define WAIT_ASYNC_0() asm volatile("s_wait_asynccnt 0" ::: "memory")
#endif
#if __has_builtin(__builtin_amdgcn_s_wait_dscnt)
#define WAIT_DS_0() __builtin_amdgcn_s_wait_dscnt(0)
#else
#define WAIT_DS_0() asm volatile("s_wait_dscnt 0" ::: "memory")
#endif
#endif // USE_ASYNC
// ---------------------------------------------------------------------------

__global__ __launch_bounds__(NTH)
void detect_nms_kernel(const float* __restrict__ loc,
                       const float* __restrict__ conf,
                       float* __restrict__ out) {
  const int t  = threadIdx.x;           // 0..1023
  const int bc = blockIdx.x;            // 0..655
  const int b  = bc / C_;
  const int c  = bc % C_;

  float* outrow = out + (size_t)bc * TOPK_ * 5;

  // Background class: all zeros.
  if (c == 0) {
    for (int j = t; j < TOPK_ * 5; j += NTH) outrow[j] = 0.0f;
    return;
  }

  const float4* loc4 = (const float4*)loc;
  const float4* gbox = loc4 + (size_t)b * P_;

  __shared__ float  red_s[32];
  __shared__ int    red_p[32];
  __shared__ float  sh_bs;
  __shared__ int    sh_bp;
  __shared__ float4 sh_bb;
  __shared__ float4 sh_pad;
  __shared__ int    sh_firstp;
#if USE_ASYNC
  __shared__ float4 stage[2][NTH];      // 32 KB double buffer
#endif

  // ---- Phase 0: load scores, threshold-mask, find first valid index ----
  float s[K_];
  int firstp = IMAX;
#pragma unroll
  for (int i = 0; i < K_; ++i) {
    const int p = i * NTH + t;
    const float sc = conf[((size_t)b * P_ + p) * C_ + c];
    const bool m = sc > CONF_T_;
    s[i] = m ? sc : NEGF;
    if (m && p < firstp) firstp = p;
  }

  if (t == 0) sh_firstp = IMAX;
  __syncthreads();
  if (firstp != IMAX) atomicMin(&sh_firstp, firstp);
  __syncthreads();
  const int  fp  = sh_firstp;
  const bool any = (fp != IMAX);
  if (t == 0) sh_pad = gbox[any ? fp : 0];
  __syncthreads();
  const float4 pad = sh_pad;

  if (!any) {                           // whole lane empty -> all-zero rows
    for (int j = t; j < TOPK_ * 5; j += NTH) outrow[j] = 0.0f;
    return;
  }

  // ---- NMS: 64 rounds of argmax + IoU suppression ----
  for (int k = 0; k < TOPK_; ++k) {
    // per-thread scan (strict '>' keeps lowest index on ties)
    float bs = NEGF;
    int   bp = IMAX;
#pragma unroll
    for (int i = 0; i < K_; ++i) {
      if (s[i] > bs) { bs = s[i]; bp = i * NTH + t; }
    }
    // wave32 reduction
    for (int off = 16; off > 0; off >>= 1) {
      const float os = __shfl_down(bs, off, 32);
      const int   op = __shfl_down(bp, off, 32);
      if (os > bs || (os == bs && op < bp)) { bs = os; bp = op; }
    }
    const int wave = t >> 5, lane = t & 31;
    if (lane == 0) { red_s[wave] = bs; red_p[wave] = bp; }
    __syncthreads();
    if (t < 32) {                       // wave 0 reduces 32 partials
      bs = red_s[t]; bp = red_p[t];
      for (int off = 16; off > 0; off >>= 1) {
        const float os = __shfl_down(bs, off, 32);
        const int   op = __shfl_down(bp, off, 32);
        if (os > bs || (os == bs && op < bp)) { bs = os; bp = op; }
      }
      if (t == 0) {
        sh_bs = bs; sh_bp = bp;
        sh_bb = (bs > NEGF && bp < P_) ? gbox[bp] : pad;
      }
    }
    __syncthreads();
    const float  vbs  = sh_bs;
    const int    vbp  = sh_bp;
    const float4 bb   = sh_bb;
    const bool   valid = (vbs > NEGF);

    if (!valid) {                       // uniform: pad out remaining rows
      const int rem = (TOPK_ - k) * 5;
      for (int j = t; j < rem; j += NTH) {
        const int kk = k + j / 5, f = j % 5;
        float v = 0.0f;
        if (f == 1) v = pad.x; else if (f == 2) v = pad.y;
        else if (f == 3) v = pad.z; else if (f == 4) v = pad.w;
        outrow[kk * 5 + f] = v;
      }
      return;
    }

    if (t == 0) {
      outrow[k * 5 + 0] = vbs;
      outrow[k * 5 + 1] = bb.x;
      outrow[k * 5 + 2] = bb.y;
      outrow[k * 5 + 3] = bb.z;
      outrow[k * 5 + 4] = bb.w;
    }

    const float areab = (bb.z - bb.x) * (bb.w - bb.y);

#if USE_ASYNC
    // Double-buffered async global->LDS staging of the box stream.
    // Each lane stages exactly the 16B it will read back (stage[buf][t]),
    // so wave-local s_wait_asynccnt is sufficient (no block barrier needed).
    async_b128(gbox + 0 * NTH + t, &stage[0][t]);
    async_b128(gbox + 1 * NTH + t, &stage[1][t]);
#pragma unroll 4
    for (int i = 0; i < K_; ++i) {
      if (i < K_ - 1) WAIT_ASYNC_1(); else WAIT_ASYNC_0();
      const float4 q = stage[i & 1][t];
      const int p = i * NTH + t;
      if (s[i] > NEGF) {
        const float xx1 = fmaxf(bb.x, q.x);
        const float yy1 = fmaxf(bb.y, q.y);
        const float xx2 = fminf(bb.z, q.z);
        const float yy2 = fminf(bb.w, q.w);
        const float inter = fmaxf(xx2 - xx1, 0.0f) * fmaxf(yy2 - yy1, 0.0f);
        const float area  = (q.z - q.x) * (q.w - q.y);
        const float iou   = inter / (area + areab - inter);
        if (iou > NMS_T_) s[i] = NEGF;
      }
      if (p == vbp) s[i] = NEGF;
      if (i + 2 < K_) {
        WAIT_DS_0();                    // LDS reads of this buffer retired
        async_b128(gbox + (i + 2) * NTH + t, &stage[i & 1][t]);
      }
    }
#else
#pragma unroll 4
    for (int i = 0; i < K_; ++i) {
      const int p = i * NTH + t;
      if (s[i] > NEGF) {
        const float4 q = gbox[p];
        const float xx1 = fmaxf(bb.x, q.x);
        const float yy1 = fmaxf(bb.y, q.y);
        const float xx2 = fminf(bb.z, q.z);
        const float yy2 = fminf(bb.w, q.w);
        const float inter = fmaxf(xx2 - xx1, 0.0f) * fmaxf(yy2 - yy1, 0.0f);
        const float area  = (q.z - q.x) * (q.w - q.y);
        const float iou   = inter / (area + areab - inter);
        if (iou > NMS_T_) s[i] = NEGF;
      }
      if (p == vbp) s[i] = NEGF;
    }
#endif
    __syncthreads();
  }
}

extern "C" void kernel_launch(void* const* d_in, const int* in_sizes, int n_in,
                              void* d_out, int out_size, void* d_ws, size_t ws_size,
                              hipStream_t stream) {
  const float* loc  = (const float*)d_in[0];   // (B,P,4) f32
  const float* conf = (const float*)d_in[1];   // (B,P,C) f32
  // d_in[2] = prior_data: unused by the reference computation.
  float* out = (float*)d_out;                  // (B,C,TOP_K,5) f32

  dim3 grid(B_ * C_);
  dim3 block(NTH);
  detect_nms_kernel<<<grid, block, 0, stream>>>(loc, conf, out);
}